// GNN_n_50414326120717
// MI455X (gfx1250) — compile-verified
//
#include <hip/hip_runtime.h>
#include <math.h>

typedef float v2f __attribute__((ext_vector_type(2)));
typedef float v8f __attribute__((ext_vector_type(8)));

#define DIM_IN  200
#define DIM_HID 100
#define DIM_H2  50
#define DIM_H2P 52    // 50 padded to multiple of 4 (A-operand row stride)
#define DIM_CLS 11
#define MT 5          // M-tiles per wave; 50000 / (16*5) = 625 exactly

// ---------------- degree / normalization ----------------
__global__ void k_init_deg(float* __restrict__ deg, int n) {
  int i = blockIdx.x * blockDim.x + threadIdx.x;
  if (i < n) deg[i] = 1.0f;  // self-loop weight
}

__global__ void k_accum_deg(const int* __restrict__ dst, const float* __restrict__ ew,
                            float* __restrict__ deg, int E) {
  int i = blockIdx.x * blockDim.x + threadIdx.x;
  if (i < E) unsafeAtomicAdd(&deg[dst[i]], ew[i]);
}

__global__ void k_make_dinv(float* __restrict__ deg, int n) {
  int i = blockIdx.x * blockDim.x + threadIdx.x;
  if (i < n) { float d = deg[i]; deg[i] = (d > 0.f) ? rsqrtf(d) : 0.f; }
}

// ---------------- weight pack: W[K][N] -> BT[Npad][K4], transposed + 0-pad --
__global__ void k_pack_w(const float* __restrict__ W, float* __restrict__ BT,
                         int K, int N, int K4, int Npad) {
  int i = blockIdx.x * blockDim.x + threadIdx.x;
  if (i < Npad * K4) {
    int n = i / K4, k = i % K4;
    BT[i] = (n < N && k < K) ? W[(size_t)k * N + n] : 0.f;
  }
}

// ---------------- fp32 WMMA GEMM (software pipelined, unroll-1) ----------
// C[M,Nout] (row stride ldc) = A[M,K4] (row stride lda=K4, zero-padded cols)
//                              @ B (packed BT[Npad][K4])
// One wave computes MT=5 consecutive 16x16 tiles down M sharing one B
// fragment. Double-buffered, branch-free K loop (#pragma unroll 1): each
// iteration issues 6 global_load_b64 (next fragments) then 5 back-to-back
// WMMAs on the previous fragments, so loads overlap the XDL ops. Named
// scalar pointers (not a pointer array) keep addrspace(1) inference ->
// global_load_b64, not flat_load (no DScnt coupling).
// fuse: 0 = none, 1 = +bias, 2 = +bias+relu. Pad cols (>=Nout, <ldc) get 0.
__global__ void k_wmma_gemm(const float* __restrict__ A, int lda,
                            const float* __restrict__ BT,
                            const float* __restrict__ bias,
                            float* __restrict__ C, int ldc,
                            int M, int Nout, int K4, int Npad, int fuse) {
  const int lane   = threadIdx.x & 31;
  const int wave   = threadIdx.x >> 5;
  const int tilesN = Npad >> 4;
  const int groups = M / (16 * MT);
  const int task   = blockIdx.x * (blockDim.x >> 5) + wave;
  if (task >= groups * tilesN) return;          // wave-uniform
  const int m0 = (task / tilesN) * (16 * MT);
  const int n0 = (task % tilesN) << 4;

  const int l16  = lane & 15;
  const int half = lane >> 4;
  const int col  = n0 + l16;
  const int koff = 2 * half;                    // A/B VGPR0: K=2*half, VGPR1: +1

  // running fragment pointers (named scalars -> stay global address space)
  const float* a0p = A + (size_t)(m0 +  0 + l16) * lda + koff;
  const float* a1p = A + (size_t)(m0 + 16 + l16) * lda + koff;
  const float* a2p = A + (size_t)(m0 + 32 + l16) * lda + koff;
  const float* a3p = A + (size_t)(m0 + 48 + l16) * lda + koff;
  const float* a4p = A + (size_t)(m0 + 64 + l16) * lda + koff;
  const float* bp  = BT + (size_t)col * K4 + koff;

  v8f acc0 = {}, acc1 = {}, acc2 = {}, acc3 = {}, acc4 = {};

  // prologue: fragments for k = 0
  v2f bq  = *(const v2f*)bp;
  v2f aq0 = *(const v2f*)a0p;
  v2f aq1 = *(const v2f*)a1p;
  v2f aq2 = *(const v2f*)a2p;
  v2f aq3 = *(const v2f*)a3p;
  v2f aq4 = *(const v2f*)a4p;

  const int steps = (K4 >> 2) - 1;
#pragma unroll 1
  for (int s = 0; s < steps; ++s) {
    // issue next-step loads first (in flight while the WMMAs execute)
    bp  += 4; const v2f bn  = *(const v2f*)bp;
    a0p += 4; const v2f an0 = *(const v2f*)a0p;
    a1p += 4; const v2f an1 = *(const v2f*)a1p;
    a2p += 4; const v2f an2 = *(const v2f*)a2p;
    a3p += 4; const v2f an3 = *(const v2f*)a3p;
    a4p += 4; const v2f an4 = *(const v2f*)a4p;
    // consume previous fragments: 5 back-to-back WMMAs
    acc0 = __builtin_amdgcn_wmma_f32_16x16x4_f32(false, aq0, false, bq, (short)0, acc0, false, false);
    acc1 = __builtin_amdgcn_wmma_f32_16x16x4_f32(false, aq1, false, bq, (short)0, acc1, false, false);
    acc2 = __builtin_amdgcn_wmma_f32_16x16x4_f32(false, aq2, false, bq, (short)0, acc2, false, false);
    acc3 = __builtin_amdgcn_wmma_f32_16x16x4_f32(false, aq3, false, bq, (short)0, acc3, false, false);
    acc4 = __builtin_amdgcn_wmma_f32_16x16x4_f32(false, aq4, false, bq, (short)0, acc4, false, false);
    bq = bn; aq0 = an0; aq1 = an1; aq2 = an2; aq3 = an3; aq4 = an4;
  }
  // epilogue WMMAs for the last k-step
  acc0 = __builtin_amdgcn_wmma_f32_16x16x4_f32(false, aq0, false, bq, (short)0, acc0, false, false);
  acc1 = __builtin_amdgcn_wmma_f32_16x16x4_f32(false, aq1, false, bq, (short)0, acc1, false, false);
  acc2 = __builtin_amdgcn_wmma_f32_16x16x4_f32(false, aq2, false, bq, (short)0, acc2, false, false);
  acc3 = __builtin_amdgcn_wmma_f32_16x16x4_f32(false, aq3, false, bq, (short)0, acc3, false, false);
  acc4 = __builtin_amdgcn_wmma_f32_16x16x4_f32(false, aq4, false, bq, (short)0, acc4, false, false);

  float bv = 0.f;
  if (fuse != 0 && col < Nout) bv = bias[col];
  const bool cvalid = (col < Nout);
  const bool cstore = (col < ldc);
  const v8f accs[MT] = {acc0, acc1, acc2, acc3, acc4};
#pragma unroll
  for (int t = 0; t < MT; ++t)
    for (int r = 0; r < 8; ++r) {
      const int row = m0 + t * 16 + r + 8 * half;   // C/D: VGPR r -> M=r+8*half
      float v = accs[t][r] + bv;
      if (fuse == 2) v = fmaxf(v, 0.f);
      v = cvalid ? v : 0.f;                          // zero the pad columns
      if (cstore) C[(size_t)row * ldc + col] = v;
    }
}

// ---------------- GCN aggregation ----------------
// agg[n,f] = h[n,f] * dinv[n]^2   (self-loop term; fully initializes agg)
__global__ void k_self_loop(const float* __restrict__ h, const float* __restrict__ dinv,
                            float* __restrict__ agg, int total, int F) {
  int i = blockIdx.x * blockDim.x + threadIdx.x;
  if (i < total) {
    float di = dinv[i / F];
    agg[i] = h[i] * di * di;
  }
}

// one wave32 per edge; lanes stride features; coef loads wave-uniform;
// scatter-add via native f32 atomics (L2-resident: 20MB << 192MB L2)
__global__ void k_scatter(const float* __restrict__ h, const int* __restrict__ src,
                          const int* __restrict__ dst, const float* __restrict__ ew,
                          const float* __restrict__ dinv, float* __restrict__ agg,
                          int E, int F) {
  const int wid  = (int)((blockIdx.x * blockDim.x + threadIdx.x) >> 5);
  const int lane = threadIdx.x & 31;
  if (wid >= E) return;                         // wave-uniform
  const int s = src[wid], d = dst[wid];
  const float coef = dinv[s] * ew[wid] * dinv[d];
  const float* hs = h + (size_t)s * F;
  float* od = agg + (size_t)d * F;
  for (int f = lane; f < F; f += 32)
    unsafeAtomicAdd(&od[f], hs[f] * coef);
}

__global__ void k_bias_relu(float* __restrict__ io, const float* __restrict__ bias,
                            int total, int F) {
  int i = blockIdx.x * blockDim.x + threadIdx.x;
  if (i < total) io[i] = fmaxf(io[i] + bias[i % F], 0.f);
}

// ---------------- log_softmax over 11 classes, in place ----------------
__global__ void k_log_softmax(float* __restrict__ io, int n) {
  int i = blockIdx.x * blockDim.x + threadIdx.x;
  if (i >= n) return;
  float* p = io + (size_t)i * DIM_CLS;
  float v[DIM_CLS];
  float m = -3.4e38f;
  for (int c = 0; c < DIM_CLS; ++c) { v[c] = p[c]; m = fmaxf(m, v[c]); }
  float s = 0.f;
  for (int c = 0; c < DIM_CLS; ++c) s += expf(v[c] - m);
  const float ls = m + logf(s);
  for (int c = 0; c < DIM_CLS; ++c) p[c] = v[c] - ls;
}

// ---------------- host launcher ----------------
static inline int ceil_div(int a, int b) { return (a + b - 1) / b; }

extern "C" void kernel_launch(void* const* d_in, const int* in_sizes, int n_in,
                              void* d_out, int out_size, void* d_ws, size_t ws_size,
                              hipStream_t stream) {
  const float* x   = (const float*)d_in[0];
  const int*   ei  = (const int*)  d_in[1];
  const float* ea  = (const float*)d_in[2];
  const float* W1  = (const float*)d_in[3];
  const float* b1  = (const float*)d_in[4];
  const float* W2  = (const float*)d_in[5];
  const float* b2  = (const float*)d_in[6];
  const float* W3  = (const float*)d_in[7];
  const float* b3  = (const float*)d_in[8];
  const float* Wm1 = (const float*)d_in[9];
  const float* bm1 = (const float*)d_in[10];
  const float* Wm2 = (const float*)d_in[11];
  const float* bm2 = (const float*)d_in[12];
  const float* Wm3 = (const float*)d_in[13];
  const float* bm3 = (const float*)d_in[14];
  float* out = (float*)d_out;

  const int N = in_sizes[0] / DIM_IN;   // 50000
  const int E = in_sizes[2];            // 800000
  const int* src = ei;
  const int* dst = ei + E;

  // padded B-pack geometries
  const int NP_HID = 112;   // ceil16(100)
  const int NP_H2  = 64;    // ceil16(50)
  const int NP_CLS = 16;    // ceil16(11)

  // workspace layout
  float* dinv = (float*)d_ws;
  float* p    = dinv + ((N + 255) & ~255);
  float* BT1  = p;  p += NP_HID * DIM_IN;    // [112][200]
  float* BT2  = p;  p += NP_HID * DIM_HID;   // [112][100]
  float* BT3  = p;  p += NP_HID * DIM_HID;   // [112][100]
  float* BTm1 = p;  p += NP_H2  * DIM_HID;   // [64][100]
  float* BTm2 = p;  p += NP_H2  * DIM_H2P;   // [64][52]
  float* BTm3 = p;  p += NP_CLS * DIM_H2P;   // [16][52]
  float* bufA = p;  p += (size_t)N * DIM_HID;
  float* bufB = p;

  const int BT = 256;

  // pack all weights (transposed, zero-padded)
  auto pack = [&](const float* W, float* Bp, int K, int Nw, int K4, int Npad) {
    k_pack_w<<<ceil_div(Npad * K4, BT), BT, 0, stream>>>(W, Bp, K, Nw, K4, Npad);
  };
  pack(W1,  BT1,  DIM_IN,  DIM_HID, DIM_IN,  NP_HID);
  pack(W2,  BT2,  DIM_HID, DIM_HID, DIM_HID, NP_HID);
  pack(W3,  BT3,  DIM_HID, DIM_HID, DIM_HID, NP_HID);
  pack(Wm1, BTm1, DIM_HID, DIM_H2,  DIM_HID, NP_H2);
  pack(Wm2, BTm2, DIM_H2,  DIM_H2,  DIM_H2P, NP_H2);
  pack(Wm3, BTm3, DIM_H2,  DIM_CLS, DIM_H2P, NP_CLS);

  // normalization (structure-only, computed once)
  k_init_deg <<<ceil_div(N, BT), BT, 0, stream>>>(dinv, N);
  k_accum_deg<<<ceil_div(E, BT), BT, 0, stream>>>(dst, ea, dinv, E);
  k_make_dinv<<<ceil_div(N, BT), BT, 0, stream>>>(dinv, N);

  auto gemm = [&](const float* A, int lda, const float* Bp, const float* bias,
                  float* C, int ldc, int Nout, int K4, int Npad, int fuse) {
    const int tasks = (N / (16 * MT)) * (Npad >> 4);
    k_wmma_gemm<<<ceil_div(tasks, 8), 256, 0, stream>>>(
        A, lda, Bp, bias, C, ldc, N, Nout, K4, Npad, fuse);
  };
  auto gcn_agg = [&](const float* h, float* agg, const float* bias, int F) {
    k_self_loop<<<ceil_div(N * F, BT), BT, 0, stream>>>(h, dinv, agg, N * F, F);
    k_scatter  <<<ceil_div(E, BT / 32), BT, 0, stream>>>(h, src, dst, ea, dinv, agg, E, F);
    k_bias_relu<<<ceil_div(N * F, BT), BT, 0, stream>>>(agg, bias, N * F, F);
  };

  // GCN layer 1: x [N,200] @ W1 -> bufA [N,100]; aggregate -> bufB
  gemm(x, DIM_IN, BT1, nullptr, bufA, DIM_HID, DIM_HID, DIM_IN, NP_HID, 0);
  gcn_agg(bufA, bufB, b1, DIM_HID);
  // GCN layer 2
  gemm(bufB, DIM_HID, BT2, nullptr, bufA, DIM_HID, DIM_HID, DIM_HID, NP_HID, 0);
  gcn_agg(bufA, bufB, b2, DIM_HID);
  // GCN layer 3
  gemm(bufB, DIM_HID, BT3, nullptr, bufA, DIM_HID, DIM_HID, DIM_HID, NP_HID, 0);
  gcn_agg(bufA, bufB, b3, DIM_HID);
  // MLP (50-wide activations stored padded to stride 52, pad cols zeroed)
  gemm(bufB, DIM_HID, BTm1, bm1, bufA, DIM_H2P, DIM_H2, DIM_HID, NP_H2, 2);
  gemm(bufA, DIM_H2P, BTm2, bm2, bufB, DIM_H2P, DIM_H2, DIM_H2P, NP_H2, 2);
  gemm(bufB, DIM_H2P, BTm3, bm3, out, DIM_CLS, DIM_CLS, DIM_H2P, NP_CLS, 1);
  k_log_softmax<<<ceil_div(N, BT), BT, 0, stream>>>(out, N);

  (void)n_in; (void)out_size; (void)ws_size;
}